// MultiHeadSelfAttention_86663850099411
// MI455X (gfx1250) — compile-verified
//
#include <hip/hip_runtime.h>
#include <hip/hip_bf16.h>
#include <math.h>
#include <stdint.h>

// ---------------------------------------------------------------------------
// Problem constants (from reference): B=4, C=256, H=W=48 -> L=2304, NH=8, DK=64, D=512
// ---------------------------------------------------------------------------
#define BB   4
#define CC   256
#define LL   2304
#define NHH  8
#define DKK  64
#define DD   512

typedef __attribute__((ext_vector_type(2))) float v2f;
typedef __attribute__((ext_vector_type(8))) float v8f;

// D = A(16x4) * B(4x16) + C, full fp32 (V_WMMA_F32_16X16X4_F32)
__device__ __forceinline__ v8f wmma_f32(v2f a, v2f b, v8f c) {
    return __builtin_amdgcn_wmma_f32_16x16x4_f32(false, a, false, b, (short)0, c, false, false);
}

// ---------------------------------------------------------------------------
// Weight transpose: in is MxN row-major, out is NxM row-major
// ---------------------------------------------------------------------------
__global__ __launch_bounds__(256) void transpose_kernel(const float* __restrict__ in,
                                                        float* __restrict__ out,
                                                        int M, int N) {
    int idx = blockIdx.x * 256 + threadIdx.x;
    if (idx < M * N) {
        int m = idx / N, n = idx - m * N;
        out[n * M + m] = in[idx];
    }
}

// ---------------------------------------------------------------------------
// QKV projection: out[b,h,l,dk] = sum_c W[d,c] * x[b,c,l]   (d = h*64+dk)
// GEMM per (b,proj): M=L (l), N=D (d), K=C.  A[l,c]=x[b,c,l] (lane-coalesced),
// B[c,d]=WT[c*D+d] (lane-coalesced).  8 waves/WG along M, each wave 16(M)x64(N).
// ---------------------------------------------------------------------------
__global__ __launch_bounds__(256) void qkv_proj_kernel(
    const float* __restrict__ x,
    const float* __restrict__ wqT, const float* __restrict__ wkT, const float* __restrict__ wvT,
    float* __restrict__ Q, float* __restrict__ K, float* __restrict__ V) {
    int bz = blockIdx.z;
    int b = bz / 3, p = bz - 3 * b;
    const float* WT = (p == 0) ? wqT : (p == 1) ? wkT : wvT;
    float* out = (p == 0) ? Q : (p == 1) ? K : V;

    int wave = threadIdx.x >> 5;
    int lane = threadIdx.x & 31;
    int l15  = lane & 15;
    int koff = (lane >> 4) * 2;   // A/B frag K offset for this lane half
    int half = lane >> 4;

    int m0 = blockIdx.x * 128 + wave * 16;  // l base
    int n0 = blockIdx.y * 64;               // d base (one head: n0 multiple of 64)
    const float* xb = x + (size_t)b * CC * LL;

    v8f acc[4] = {};
    for (int k = 0; k < CC; k += 4) {
        v2f a;
        a.x = xb[(size_t)(k + koff) * LL + m0 + l15];
        a.y = xb[(size_t)(k + koff + 1) * LL + m0 + l15];
        const float* w0 = WT + (size_t)(k + koff) * DD + n0;
        const float* w1 = w0 + DD;
#pragma unroll
        for (int j = 0; j < 4; ++j) {
            v2f bb;
            bb.x = w0[j * 16 + l15];
            bb.y = w1[j * 16 + l15];
            acc[j] = wmma_f32(a, bb, acc[j]);
        }
    }
    // store into [B][NH][L][DK]
    int h = n0 >> 6;
    float* op = out + ((size_t)(b * NHH + h) * LL + m0 + half * 8) * DKK;
#pragma unroll
    for (int j = 0; j < 4; ++j)
#pragma unroll
        for (int i = 0; i < 8; ++i)
            op[(size_t)i * DKK + j * 16 + l15] = acc[j][i];
}

// ---------------------------------------------------------------------------
// Flash attention. One WG = 128 queries of one (b,h); 8 waves x 16 queries.
// Streams 16-key blocks with DOUBLE-BUFFERED async global->LDS copies
// (GLOBAL_LOAD_ASYNC_TO_LDS_B128, ASYNCcnt): the next block's K/V tiles are
// in flight while WMMAs consume the current one. S=Q*K^T via WMMA, online
// softmax (16-lane shuffle row reductions), P staged via per-wave LDS tile
// into A-fragments, O += P*V via WMMA. Output AO[b,l,d] coalesced.
// ---------------------------------------------------------------------------
__global__ __launch_bounds__(256) void attn_kernel(
    const float* __restrict__ Q, const float* __restrict__ K, const float* __restrict__ V,
    float* __restrict__ AO) {
    // pad 76 floats/row: rows 16B-aligned for b128 async writes; frag reads at
    // lane-stride 76 -> banks (12n mod 64), all 16 distinct -> conflict-free.
    __shared__ __align__(16) float Kt[2][16][76];
    __shared__ __align__(16) float Vt[2][16][76];
    __shared__ __align__(16) float Pt[8][16][17];   // per-wave P staging (C/D -> A layout)

    int bh = blockIdx.y;                 // b*8 + h
    int b = bh >> 3, h = bh & 7;
    int wave = threadIdx.x >> 5;
    int lane = threadIdx.x & 31;
    int l15  = lane & 15;
    int koff = (lane >> 4) * 2;
    int half = lane >> 4;

    int q0 = blockIdx.x * 128 + wave * 16;
    const float* Qb = Q + (size_t)bh * LL * DKK;
    const float* Kb = K + (size_t)bh * LL * DKK;
    const float* Vb = V + (size_t)bh * LL * DKK;

    // per-thread slot for cooperative K/V tile copies (16B each)
    int r  = threadIdx.x >> 4;
    int c4 = (threadIdx.x & 15) * 4;

    // issue async global->LDS copies for key-block t into buffer nb
    auto issue_async = [&](int t, int nb) {
        const float* gk = Kb + (size_t)(t * 16 + r) * DKK + c4;
        const float* gv = Vb + (size_t)(t * 16 + r) * DKK + c4;
        unsigned lk = (unsigned)(uintptr_t)&Kt[nb][r][c4];  // low 32 bits = LDS offset
        unsigned lv = (unsigned)(uintptr_t)&Vt[nb][r][c4];
        asm volatile("global_load_async_to_lds_b128 %0, %1, off"
                     :: "v"(lk), "v"(gk) : "memory");
        asm volatile("global_load_async_to_lds_b128 %0, %1, off"
                     :: "v"(lv), "v"(gv) : "memory");
    };

    // Q A-fragments for all 16 K-steps, pre-scaled by 1/sqrt(DK)=0.125
    v2f qa[16];
#pragma unroll
    for (int kk = 0; kk < 16; ++kk) {
        const float* qp = Qb + (size_t)(q0 + l15) * DKK + kk * 4 + koff;
        qa[kk].x = qp[0] * 0.125f;
        qa[kk].y = qp[1] * 0.125f;
    }

    v8f o[4] = {};
    float rm[8], rl[8];
#pragma unroll
    for (int i = 0; i < 8; ++i) { rm[i] = -__builtin_inff(); rl[i] = 0.0f; }

    float (*Pw)[17] = Pt[wave];
    const int NT = LL / 16;

    issue_async(0, 0);   // prologue: block 0 -> buffer 0

    for (int t = 0; t < NT; ++t) {
        int buf = t & 1;
        if (t + 1 < NT) {
            issue_async(t + 1, buf ^ 1);               // prefetch next block
            asm volatile("s_wait_asynccnt 0x2" ::: "memory");  // block t's 2 ops done
        } else {
            asm volatile("s_wait_asynccnt 0x0" ::: "memory");
        }
        __syncthreads();   // all waves' copies of `buf` have landed

        // S = (Q/8) * K^T : 16 WMMA over K-dim 64
        v8f s = {};
#pragma unroll
        for (int kk = 0; kk < 16; ++kk) {
            v2f bb;
            bb.x = Kt[buf][l15][kk * 4 + koff];
            bb.y = Kt[buf][l15][kk * 4 + koff + 1];
            s = wmma_f32(qa[kk], bb, s);
        }

        // online softmax update; row i lives entirely in one 16-lane half
#pragma unroll
        for (int i = 0; i < 8; ++i) {
            float xv = s[i];
            float mx = xv;
#pragma unroll
            for (int off = 1; off < 16; off <<= 1)
                mx = fmaxf(mx, __shfl_xor(mx, off, 16));
            float mnew = fmaxf(rm[i], mx);
            float al   = __expf(rm[i] - mnew);
            float e    = __expf(xv - mnew);
            float sum  = e;
#pragma unroll
            for (int off = 1; off < 16; off <<= 1)
                sum += __shfl_xor(sum, off, 16);
            rl[i] = rl[i] * al + sum;
            rm[i] = mnew;
            o[0][i] *= al; o[1][i] *= al; o[2][i] *= al; o[3][i] *= al;
            Pw[i + 8 * half][l15] = e;   // C/D layout -> LDS [m][key]
        }

        // O += P * V : K-dim 16 -> 4 steps x 4 N-groups
#pragma unroll
        for (int kk = 0; kk < 4; ++kk) {
            v2f pa;
            pa.x = Pw[l15][kk * 4 + koff];
            pa.y = Pw[l15][kk * 4 + koff + 1];
#pragma unroll
            for (int j = 0; j < 4; ++j) {
                v2f vb;
                vb.x = Vt[buf][kk * 4 + koff][j * 16 + l15];
                vb.y = Vt[buf][kk * 4 + koff + 1][j * 16 + l15];
                o[j] = wmma_f32(pa, vb, o[j]);
            }
        }
        __syncthreads();   // drains DScnt too: `buf` free for the t+1 prefetch
    }

    // normalize and store AO[b, l, d] (d contiguous -> coalesced)
    float inv[8];
#pragma unroll
    for (int i = 0; i < 8; ++i) inv[i] = 1.0f / rl[i];
    float* ao = AO + ((size_t)b * LL + q0 + half * 8) * DD + h * DKK;
#pragma unroll
    for (int j = 0; j < 4; ++j)
#pragma unroll
        for (int i = 0; i < 8; ++i)
            ao[(size_t)i * DD + j * 16 + l15] = o[j][i] * inv[i];
}

// ---------------------------------------------------------------------------
// Output projection: Y[b,c,l] = sum_d Wo[c,d] * AO[b,l,d]
// M=C (c), N=L (l), K=D.  A[c,d]=WoT[d*C+c] (coalesced),
// B[d,l]=AO[b,l,d] staged through LDS (coalesced global, padded LDS reads).
// WG = 8 waves along c (128 c) x 64 l.
// ---------------------------------------------------------------------------
__global__ __launch_bounds__(256) void oproj_kernel(
    const float* __restrict__ AO, const float* __restrict__ WoT, float* __restrict__ Y) {
    __shared__ __align__(16) float At[64][36];   // [l][d-chunk 32], pad 4 (row 144B, banks 4n)

    int b  = blockIdx.z;
    int l0 = blockIdx.x * 64;
    int c0 = blockIdx.y * 128;
    int wave = threadIdx.x >> 5;
    int lane = threadIdx.x & 31;
    int l15  = lane & 15;
    int koff = (lane >> 4) * 2;
    int half = lane >> 4;
    int cw = c0 + wave * 16;

    v8f acc[4] = {};
    for (int d0 = 0; d0 < DD; d0 += 32) {
        __syncthreads();
#pragma unroll
        for (int r = 0; r < 2; ++r) {    // 512 float4 loads, 2 per thread
            int idx = threadIdx.x * 2 + r;
            int ll = idx >> 3, d4 = (idx & 7) * 4;
            float4 v = *(const float4*)(AO + ((size_t)b * LL + l0 + ll) * DD + d0 + d4);
            *(float4*)&At[ll][d4] = v;
        }
        __syncthreads();
#pragma unroll
        for (int kk = 0; kk < 8; ++kk) {
            int kd = kk * 4 + koff;
            const float* wp = WoT + (size_t)(d0 + kd) * CC + cw + l15;
            v2f a;
            a.x = wp[0];
            a.y = wp[CC];
#pragma unroll
            for (int j = 0; j < 4; ++j) {
                v2f bb;
                bb.x = At[j * 16 + l15][kd];
                bb.y = At[j * 16 + l15][kd + 1];
                acc[j] = wmma_f32(a, bb, acc[j]);
            }
        }
    }
    // store Y[b][c][l], lanes along l -> coalesced
#pragma unroll
    for (int j = 0; j < 4; ++j)
#pragma unroll
        for (int i = 0; i < 8; ++i)
            Y[((size_t)b * CC + cw + half * 8 + i) * LL + l0 + j * 16 + l15] = acc[j][i];
}

// ---------------------------------------------------------------------------
// BatchNorm statistics: one workgroup per channel, reduce over B*L elements
// stats[c] = mean, stats[C+c] = rsqrt(var + eps)
// ---------------------------------------------------------------------------
__global__ __launch_bounds__(256) void bn_stats_kernel(const float* __restrict__ Y,
                                                       float* __restrict__ stats) {
    int c = blockIdx.x;
    float s = 0.0f, s2 = 0.0f;
    for (int b = 0; b < BB; ++b) {
        const float* p = Y + ((size_t)b * CC + c) * LL;
        for (int l = threadIdx.x; l < LL; l += 256) {
            float v = p[l];
            s += v; s2 += v * v;
        }
    }
    __shared__ float sh0[256], sh1[256];
    sh0[threadIdx.x] = s; sh1[threadIdx.x] = s2;
    __syncthreads();
    for (int off = 128; off > 0; off >>= 1) {
        if (threadIdx.x < off) {
            sh0[threadIdx.x] += sh0[threadIdx.x + off];
            sh1[threadIdx.x] += sh1[threadIdx.x + off];
        }
        __syncthreads();
    }
    if (threadIdx.x == 0) {
        float n = (float)(BB * LL);
        float mean = sh0[0] / n;
        float var = sh1[0] / n - mean * mean;
        stats[c] = mean;
        stats[CC + c] = rsqrtf(var + 1e-5f);
    }
}

// ---------------------------------------------------------------------------
// Fused BN apply + gamma-scale + residual + LeakyReLU
// ---------------------------------------------------------------------------
__global__ __launch_bounds__(256) void finalize_kernel(
    const float* __restrict__ Y, const float* __restrict__ X,
    const float* __restrict__ stats, const float* __restrict__ bn_w,
    const float* __restrict__ bn_b, const float* __restrict__ gamma,
    float* __restrict__ out) {
    unsigned idx = blockIdx.x * 256 + threadIdx.x;   // < B*C*L = 2359296
    unsigned c = (idx / LL) & (CC - 1);
    float mean = stats[c], rstd = stats[CC + c];
    float v = (Y[idx] - mean) * rstd * bn_w[c] + bn_b[c];
    v = gamma[0] * v + X[idx];
    out[idx] = (v >= 0.0f) ? v : 0.01f * v;
}

// ---------------------------------------------------------------------------
// Host launcher
// ---------------------------------------------------------------------------
extern "C" void kernel_launch(void* const* d_in, const int* in_sizes, int n_in,
                              void* d_out, int out_size, void* d_ws, size_t ws_size,
                              hipStream_t stream) {
    (void)in_sizes; (void)n_in; (void)out_size; (void)ws_size;
    const float* x     = (const float*)d_in[0];
    const float* Wq    = (const float*)d_in[1];
    const float* Wk    = (const float*)d_in[2];
    const float* Wv    = (const float*)d_in[3];
    const float* Wo    = (const float*)d_in[4];
    const float* bn_w  = (const float*)d_in[5];
    const float* bn_b  = (const float*)d_in[6];
    const float* gamma = (const float*)d_in[7];

    float* ws = (float*)d_ws;
    const size_t QKV_SZ = (size_t)BB * NHH * LL * DKK;   // 4,718,592
    float* Q     = ws;
    float* K     = Q + QKV_SZ;
    float* V     = K + QKV_SZ;
    float* AO    = V + QKV_SZ;                 // [B][L][D], same size
    float* Y     = AO + QKV_SZ;                // [B][C][L]
    float* WqT   = Y + (size_t)BB * CC * LL;
    float* WkT   = WqT + (size_t)DD * CC;
    float* WvT   = WkT + (size_t)DD * CC;
    float* WoT   = WvT + (size_t)DD * CC;
    float* stats = WoT + (size_t)DD * CC;

    transpose_kernel<<<512, 256, 0, stream>>>(Wq, WqT, DD, CC);   // -> [C][D]
    transpose_kernel<<<512, 256, 0, stream>>>(Wk, WkT, DD, CC);
    transpose_kernel<<<512, 256, 0, stream>>>(Wv, WvT, DD, CC);
    transpose_kernel<<<512, 256, 0, stream>>>(Wo, WoT, CC, DD);   // -> [D][C]

    qkv_proj_kernel<<<dim3(LL / 128, DD / 64, BB * 3), 256, 0, stream>>>(
        x, WqT, WkT, WvT, Q, K, V);
    attn_kernel<<<dim3(LL / 128, BB * NHH), 256, 0, stream>>>(Q, K, V, AO);
    oproj_kernel<<<dim3(LL / 64, CC / 128, BB), 256, 0, stream>>>(AO, WoT, Y);
    bn_stats_kernel<<<CC, 256, 0, stream>>>(Y, stats);
    finalize_kernel<<<(BB * CC * LL) / 256, 256, 0, stream>>>(
        Y, x, stats, bn_w, bn_b, gamma, (float*)d_out);
}